// GraphPoolingLayer_56788057588224
// MI455X (gfx1250) — compile-verified
//
#include <hip/hip_runtime.h>

// Problem constants (fixed by the reference):
//   features:   (B=8, N=65536, C=64)  fp32
//   coarse_map: (B=8, NPOINT=16384, NSAMPLE=32) int32, values in [0, N)
//   output:     (B, NPOINT, C) fp32 = max over NSAMPLE gathered rows
#define BATCH   8
#define NNODES  65536
#define CDIM    64
#define NPOINT  16384
#define NSAMPLE 32

typedef __attribute__((ext_vector_type(2))) float v2f;

// One wave32 per (b, point). Each lane owns 2 channels (float2):
// a gathered row = 32 lanes x 8B = 256B = one coalesced global_load_b64 (saddr form).
// Indices are wave-uniform -> readlane to SGPR -> scalar row-base math on the SALU.
__global__ __launch_bounds__(256) void gather_max_kernel(
    const float* __restrict__ features,
    const int*   __restrict__ coarse_map,
    float*       __restrict__ out)
{
    const int lane      = threadIdx.x & 31;
    // Force wave-uniformity so all downstream addressing is scalar.
    const int waveInBlk = __builtin_amdgcn_readfirstlane((int)(threadIdx.x >> 5));
    const int gwave     = (int)blockIdx.x * 8 + waveInBlk;   // 0 .. B*NPOINT-1
    const unsigned b    = (unsigned)gwave >> 14;             // / NPOINT
    const unsigned p    = (unsigned)gwave & (NPOINT - 1);

    // --- this point's 32 indices: one coalesced 128B load, NT (read once)
    const int* __restrict__ idxp =
        coarse_map + ((size_t)b * NPOINT + (size_t)p) * NSAMPLE;
    const int idxv = __builtin_nontemporal_load(idxp + lane);

    const v2f* __restrict__ fbase =
        reinterpret_cast<const v2f*>(features) + (size_t)b * ((size_t)NNODES * (CDIM / 2));

    v2f acc;
    acc.x = -__builtin_inff();
    acc.y = -__builtin_inff();

    // Fully unrolled, 2 samples per step: up to 32 independent 256B row loads
    // in flight per wave; pairwise max enables v_max3_f32 fusion.
#pragma unroll
    for (int s = 0; s < NSAMPLE; s += 2) {
        const unsigned i0 = (unsigned)__builtin_amdgcn_readlane(idxv, s);     // SGPR
        const unsigned i1 = (unsigned)__builtin_amdgcn_readlane(idxv, s + 1); // SGPR
        const v2f v0 = fbase[(size_t)i0 * (CDIM / 2) + lane]; // saddr + lane*8
        const v2f v1 = fbase[(size_t)i1 * (CDIM / 2) + lane]; // saddr + lane*8
        acc.x = __builtin_fmaxf(__builtin_fmaxf(v0.x, v1.x), acc.x);
        acc.y = __builtin_fmaxf(__builtin_fmaxf(v0.y, v1.y), acc.y);
    }

    // Output row written once -> nontemporal store, keep L2 for features.
    v2f* __restrict__ op =
        reinterpret_cast<v2f*>(out) + ((size_t)b * NPOINT + (size_t)p) * (CDIM / 2);
    __builtin_nontemporal_store(acc, op + lane);
}

extern "C" void kernel_launch(void* const* d_in, const int* in_sizes, int n_in,
                              void* d_out, int out_size, void* d_ws, size_t ws_size,
                              hipStream_t stream)
{
    const float* features   = (const float*)d_in[0];
    const int*   coarse_map = (const int*)d_in[1];
    float*       out        = (float*)d_out;

    // B*NPOINT waves total, 8 waves (256 threads) per block.
    const int total_waves = BATCH * NPOINT;          // 131072
    const int blocks      = total_waves / 8;         // 16384

    gather_max_kernel<<<blocks, 256, 0, stream>>>(features, coarse_map, out);
}